// ForwardDeformer_30391188586525
// MI455X (gfx1250) — compile-verified
//
#include <hip/hip_runtime.h>
#include <hip/hip_bf16.h>

#define HDIM 128
#define NJ   24
#define NIB  9
#define NWAVE 8
#define PTSW 16
#define BLK_PTS (NWAVE * PTSW)
#define SOFT_BLENDF 20.0f
#define BROYDEN_STEPS 10
#define CVG_THRESHF 1e-5f

typedef __attribute__((ext_vector_type(2))) float v2f;
typedef __attribute__((ext_vector_type(8))) float v8f;

struct Smem {
  float W2[HDIM][HDIM];        // [k][n], row-major as in memory
  float W1b[4][HDIM];          // rows 0-2 = W1[k][n], row 3 = b1 (homogeneous trick)
  float W3[HDIM][NJ];          // [k][j]
  float b2[HDIM];
  float b3[NJ];
  float Amat[NJ][12];          // tfs[j][:3][:4] row-major
  float Tinv[NIB][12];         // affine inverses of init bones
  float z1[NWAVE][PTSW][HDIM]; // h1 = softplus(z1); later sigmoid(z1)
  float z2[NWAVE][PTSW][HDIM]; // h2 = softplus(z2); later sigmoid(z2)
  float dz2[NWAVE][PTSW][HDIM];// tangent layer-2 output (raw)
  float logit[NWAVE][PTSW][NJ];
  float wjs[NWAVE][PTSW][NJ];  // softmax weights
  float dlog[NWAVE][PTSW][NJ]; // tangent logits for current channel
  float xs[NWAVE][PTSW][4];    // current x per point
};

// fast softplus: args bounded -> native v_exp_f32 / v_log_f32 are safe here
__device__ __forceinline__ float softplusf(float z) {
  return fmaxf(z, 0.0f) + __logf(1.0f + __expf(-fabsf(z)));
}

__device__ __forceinline__ v8f wmma4(v2f a, v2f b, v8f c) {
  // V_WMMA_F32_16X16X4_F32 : D(16x16,f32) = A(16x4,f32) * B(4x16,f32) + C
  return __builtin_amdgcn_wmma_f32_16x16x4_f32(false, a, false, b, (short)0, c,
                                               false, false);
}

// Layer 1: z1 = [x,1] @ [W1;b1]  -> store softplus into z1
__device__ __forceinline__ void gemm_l1(Smem& sm, int wv, int half, int lc) {
  v2f a;
  if (half == 0) { a[0] = sm.xs[wv][lc][0]; a[1] = sm.xs[wv][lc][1]; }
  else           { a[0] = sm.xs[wv][lc][2]; a[1] = 1.0f; }
  #pragma unroll
  for (int nt = 0; nt < 8; ++nt) {
    int col = nt * 16 + lc;
    v2f b; b[0] = sm.W1b[2 * half][col]; b[1] = sm.W1b[2 * half + 1][col];
    v8f c = {};
    c = wmma4(a, b, c);
    #pragma unroll
    for (int v = 0; v < 8; ++v)
      sm.z1[wv][v + half * 8][col] = softplusf(c[v]);
  }
}

// Layer 2: z2 = h1 @ W2 + b2 -> store softplus into z2
__device__ __forceinline__ void gemm_l2(Smem& sm, int wv, int half, int lc) {
  for (int nt = 0; nt < 8; ++nt) {
    int col = nt * 16 + lc;
    float bias = sm.b2[col];
    v8f c;
    #pragma unroll
    for (int v = 0; v < 8; ++v) c[v] = bias;
    #pragma unroll 4
    for (int kk = 0; kk < 32; ++kk) {
      int f0 = 4 * kk + 2 * half;
      v2f a; a[0] = sm.z1[wv][lc][f0]; a[1] = sm.z1[wv][lc][f0 + 1];
      v2f b; b[0] = sm.W2[f0][col];    b[1] = sm.W2[f0 + 1][col];
      c = wmma4(a, b, c);
    }
    #pragma unroll
    for (int v = 0; v < 8; ++v)
      sm.z2[wv][v + half * 8][col] = softplusf(c[v]);
  }
}

// Layer 3: logits = h2 @ W3 + b3 (24 cols, padded with zero selects)
__device__ __forceinline__ void gemm_l3(Smem& sm, int wv, int half, int lc) {
  #pragma unroll
  for (int nt = 0; nt < 2; ++nt) {
    int col = nt * 16 + lc;
    bool okc = col < NJ;
    float bias = okc ? sm.b3[okc ? col : 0] : 0.0f;
    v8f c;
    #pragma unroll
    for (int v = 0; v < 8; ++v) c[v] = bias;
    #pragma unroll 4
    for (int kk = 0; kk < 32; ++kk) {
      int f0 = 4 * kk + 2 * half;
      v2f a; a[0] = sm.z2[wv][lc][f0]; a[1] = sm.z2[wv][lc][f0 + 1];
      v2f b;
      b[0] = okc ? sm.W3[f0][okc ? col : 0]     : 0.0f;
      b[1] = okc ? sm.W3[f0 + 1][okc ? col : 0] : 0.0f;
      c = wmma4(a, b, c);
    }
    if (okc) {
      #pragma unroll
      for (int v = 0; v < 8; ++v)
        sm.logit[wv][v + half * 8][col] = c[v];
    }
  }
}

// Tangent layer 2: dz2 = (sig1 .* W1[ch,:]) @ W2   (z1 holds sig1)
__device__ __forceinline__ void gemm_l2t(Smem& sm, int wv, int half, int lc, int ch) {
  for (int nt = 0; nt < 8; ++nt) {
    int col = nt * 16 + lc;
    v8f c = {};
    #pragma unroll 4
    for (int kk = 0; kk < 32; ++kk) {
      int f0 = 4 * kk + 2 * half;
      v2f a;
      a[0] = sm.z1[wv][lc][f0]     * sm.W1b[ch][f0];
      a[1] = sm.z1[wv][lc][f0 + 1] * sm.W1b[ch][f0 + 1];
      v2f b; b[0] = sm.W2[f0][col]; b[1] = sm.W2[f0 + 1][col];
      c = wmma4(a, b, c);
    }
    #pragma unroll
    for (int v = 0; v < 8; ++v)
      sm.dz2[wv][v + half * 8][col] = c[v];
  }
}

// Tangent layer 3: dlog = (sig2 .* dz2) @ W3   (z2 holds sig2)
__device__ __forceinline__ void gemm_l3t(Smem& sm, int wv, int half, int lc) {
  #pragma unroll
  for (int nt = 0; nt < 2; ++nt) {
    int col = nt * 16 + lc;
    bool okc = col < NJ;
    v8f c = {};
    #pragma unroll 4
    for (int kk = 0; kk < 32; ++kk) {
      int f0 = 4 * kk + 2 * half;
      v2f a;
      a[0] = sm.z2[wv][lc][f0]     * sm.dz2[wv][lc][f0];
      a[1] = sm.z2[wv][lc][f0 + 1] * sm.dz2[wv][lc][f0 + 1];
      v2f b;
      b[0] = okc ? sm.W3[f0][okc ? col : 0]     : 0.0f;
      b[1] = okc ? sm.W3[f0 + 1][okc ? col : 0] : 0.0f;
      c = wmma4(a, b, c);
    }
    if (okc) {
      #pragma unroll
      for (int v = 0; v < 8; ++v)
        sm.dlog[wv][v + half * 8][col] = c[v];
    }
  }
}

__device__ __forceinline__ void mlp_value(Smem& sm, int wv, int half, int lc) {
  gemm_l1(sm, wv, half, lc); __syncthreads();
  gemm_l2(sm, wv, half, lc); __syncthreads();
  gemm_l3(sm, wv, half, lc); __syncthreads();
}

// softmax(20*logits) -> wjs; f = sum_j w_j * (A_j [x,1])
__device__ __forceinline__ void softmax_lbs(Smem& sm, int wv, int lc,
                                            float x0, float x1, float x2,
                                            float& f0, float& f1, float& f2) {
  float mx = -1e30f;
  for (int j = 0; j < NJ; ++j)
    mx = fmaxf(mx, SOFT_BLENDF * sm.logit[wv][lc][j]);
  float Z = 0.0f;
  for (int j = 0; j < NJ; ++j) {
    float e = __expf(SOFT_BLENDF * sm.logit[wv][lc][j] - mx);  // arg <= 0
    sm.wjs[wv][lc][j] = e;
    Z += e;
  }
  float invZ = 1.0f / Z;
  f0 = 0.0f; f1 = 0.0f; f2 = 0.0f;
  for (int j = 0; j < NJ; ++j) {
    float w = sm.wjs[wv][lc][j] * invZ;
    sm.wjs[wv][lc][j] = w;
    const float* A = sm.Amat[j];
    f0 += w * (A[0] * x0 + A[1] * x1 + A[2]  * x2 + A[3]);
    f1 += w * (A[4] * x0 + A[5] * x1 + A[6]  * x2 + A[7]);
    f2 += w * (A[8] * x0 + A[9] * x1 + A[10] * x2 + A[11]);
  }
}

__global__ __launch_bounds__(256) void deformer_broyden_kernel(
    const float* __restrict__ xd, const float* __restrict__ tfs,
    const float* __restrict__ W1, const float* __restrict__ b1,
    const float* __restrict__ W2, const float* __restrict__ b2,
    const float* __restrict__ W3, const float* __restrict__ b3,
    float* __restrict__ out, int Npts) {
  __shared__ Smem sm;
  const int tid = threadIdx.x;
  const int nthr = blockDim.x;

  // ---- stage weights & transforms in LDS ----
  for (int i = tid; i < HDIM * HDIM; i += nthr) sm.W2[i >> 7][i & 127] = W2[i];
  for (int i = tid; i < 3 * HDIM; i += nthr)    sm.W1b[i >> 7][i & 127] = W1[i];
  for (int i = tid; i < HDIM; i += nthr) { sm.W1b[3][i] = b1[i]; sm.b2[i] = b2[i]; }
  for (int i = tid; i < HDIM * NJ; i += nthr)   sm.W3[i / NJ][i % NJ] = W3[i];
  for (int i = tid; i < NJ; i += nthr)          sm.b3[i] = b3[i];
  for (int i = tid; i < NJ * 12; i += nthr)     sm.Amat[i / 12][i % 12] = tfs[(i / 12) * 16 + (i % 12)];
  if (tid < NIB) {
    const int ibtab[NIB] = {0, 1, 2, 4, 5, 16, 17, 18, 19};
    const float* T = tfs + ibtab[tid] * 16;
    float r00 = T[0], r01 = T[1], r02 = T[2],  t0 = T[3];
    float r10 = T[4], r11 = T[5], r12 = T[6],  t1 = T[7];
    float r20 = T[8], r21 = T[9], r22 = T[10], t2 = T[11];
    float det = r00 * (r11 * r22 - r12 * r21)
              - r01 * (r10 * r22 - r12 * r20)
              + r02 * (r10 * r21 - r11 * r20);
    float id = 1.0f / det;
    float i00 =  (r11 * r22 - r12 * r21) * id;
    float i01 = -(r01 * r22 - r02 * r21) * id;
    float i02 =  (r01 * r12 - r02 * r11) * id;
    float i10 = -(r10 * r22 - r12 * r20) * id;
    float i11 =  (r00 * r22 - r02 * r20) * id;
    float i12 = -(r00 * r12 - r02 * r10) * id;
    float i20 =  (r10 * r21 - r11 * r20) * id;
    float i21 = -(r00 * r21 - r01 * r20) * id;
    float i22 =  (r00 * r11 - r01 * r10) * id;
    float* Ti = sm.Tinv[tid];
    Ti[0] = i00; Ti[1] = i01; Ti[2]  = i02; Ti[3]  = -(i00 * t0 + i01 * t1 + i02 * t2);
    Ti[4] = i10; Ti[5] = i11; Ti[6]  = i12; Ti[7]  = -(i10 * t0 + i11 * t1 + i12 * t2);
    Ti[8] = i20; Ti[9] = i21; Ti[10] = i22; Ti[11] = -(i20 * t0 + i21 * t1 + i22 * t2);
  }
  __syncthreads();

  const int lane = tid & 31;
  const int wv   = tid >> 5;
  const int half = lane >> 4;
  const int lc   = lane & 15;
  const int M = Npts * NIB;
  int m = blockIdx.x * BLK_PTS + wv * PTSW + lc;
  if (m >= M) m = M - 1;  // clamp; duplicate lanes recompute same point
  const int p   = m / NIB;
  const int ibn = m % NIB;

  float t0 = 0.f, t1 = 0.f, t2 = 0.f, x0 = 0.f, x1 = 0.f, x2 = 0.f;
  if (half == 0) {
    t0 = xd[p * 3 + 0]; t1 = xd[p * 3 + 1]; t2 = xd[p * 3 + 2];
    const float* Ti = sm.Tinv[ibn];
    x0 = Ti[0] * t0 + Ti[1] * t1 + Ti[2]  * t2 + Ti[3];
    x1 = Ti[4] * t0 + Ti[5] * t1 + Ti[6]  * t2 + Ti[7];
    x2 = Ti[8] * t0 + Ti[9] * t1 + Ti[10] * t2 + Ti[11];
    sm.xs[wv][lc][0] = x0; sm.xs[wv][lc][1] = x1; sm.xs[wv][lc][2] = x2;
  }
  __syncthreads();

  // ---- value channel at x_init ----
  mlp_value(sm, wv, half, lc);
  // convert h -> sigmoid in place: sig = 1 - exp(-softplus(z))
  for (int i = lane; i < PTSW * HDIM; i += 32) {
    float h1v = sm.z1[wv][i >> 7][i & 127];
    sm.z1[wv][i >> 7][i & 127] = 1.0f - __expf(-h1v);
    float h2v = sm.z2[wv][i >> 7][i & 127];
    sm.z2[wv][i >> 7][i & 127] = 1.0f - __expf(-h2v);
  }
  float f0 = 0.f, f1 = 0.f, f2 = 0.f, g0 = 0.f, g1 = 0.f, g2 = 0.f;
  if (half == 0) {
    softmax_lbs(sm, wv, lc, x0, x1, x2, f0, f1, f2);
    g0 = f0 - t0; g1 = f1 - t1; g2 = f2 - t2;
  }
  __syncthreads();

  // ---- tangent channels: build J = df/dx, then invert -> Jinv0 ----
  float Ji[3][3];
  #pragma unroll
  for (int ch = 0; ch < 3; ++ch) {
    gemm_l2t(sm, wv, half, lc, ch); __syncthreads();
    gemm_l3t(sm, wv, half, lc);     __syncthreads();
    if (half == 0) {
      float s = 0.f, u0 = 0.f, u1 = 0.f, u2 = 0.f, r0 = 0.f, r1 = 0.f, r2 = 0.f;
      for (int j = 0; j < NJ; ++j) {
        float w = sm.wjs[wv][lc][j];
        float d = sm.dlog[wv][lc][j];
        const float* A = sm.Amat[j];
        float y0 = A[0] * x0 + A[1] * x1 + A[2]  * x2 + A[3];
        float y1 = A[4] * x0 + A[5] * x1 + A[6]  * x2 + A[7];
        float y2 = A[8] * x0 + A[9] * x1 + A[10] * x2 + A[11];
        float wd = w * d;
        s += wd; u0 += wd * y0; u1 += wd * y1; u2 += wd * y2;
        r0 += w * A[0 + ch]; r1 += w * A[4 + ch]; r2 += w * A[8 + ch];
      }
      Ji[0][ch] = SOFT_BLENDF * (u0 - s * f0) + r0;
      Ji[1][ch] = SOFT_BLENDF * (u1 - s * f1) + r1;
      Ji[2][ch] = SOFT_BLENDF * (u2 - s * f2) + r2;
    }
    __syncthreads();
  }
  if (half == 0) {
    float a00 = Ji[0][0], a01 = Ji[0][1], a02 = Ji[0][2];
    float a10 = Ji[1][0], a11 = Ji[1][1], a12 = Ji[1][2];
    float a20 = Ji[2][0], a21 = Ji[2][1], a22 = Ji[2][2];
    float det = a00 * (a11 * a22 - a12 * a21)
              - a01 * (a10 * a22 - a12 * a20)
              + a02 * (a10 * a21 - a11 * a20);
    float id = 1.0f / det;
    Ji[0][0] = (a11 * a22 - a12 * a21) * id;
    Ji[0][1] = (a02 * a21 - a01 * a22) * id;
    Ji[0][2] = (a01 * a12 - a02 * a11) * id;
    Ji[1][0] = (a12 * a20 - a10 * a22) * id;
    Ji[1][1] = (a00 * a22 - a02 * a20) * id;
    Ji[1][2] = (a02 * a10 - a00 * a12) * id;
    Ji[2][0] = (a10 * a21 - a11 * a20) * id;
    Ji[2][1] = (a01 * a20 - a00 * a21) * id;
    Ji[2][2] = (a00 * a11 - a01 * a10) * id;
  }

  float bx0 = x0, bx1 = x1, bx2 = x2;
  float berr = sqrtf(g0 * g0 + g1 * g1 + g2 * g2);

  // ---- Broyden iterations ----
  for (int it = 0; it < BROYDEN_STEPS; ++it) {
    float u0 = 0.f, u1 = 0.f, u2 = 0.f;
    if (half == 0) {
      u0 = -(Ji[0][0] * g0 + Ji[0][1] * g1 + Ji[0][2] * g2);
      u1 = -(Ji[1][0] * g0 + Ji[1][1] * g1 + Ji[1][2] * g2);
      u2 = -(Ji[2][0] * g0 + Ji[2][1] * g1 + Ji[2][2] * g2);
      x0 += u0; x1 += u1; x2 += u2;
      sm.xs[wv][lc][0] = x0; sm.xs[wv][lc][1] = x1; sm.xs[wv][lc][2] = x2;
    }
    __syncthreads();
    mlp_value(sm, wv, half, lc);
    if (half == 0) {
      float nf0, nf1, nf2;
      softmax_lbs(sm, wv, lc, x0, x1, x2, nf0, nf1, nf2);
      float ng0 = nf0 - t0, ng1 = nf1 - t1, ng2 = nf2 - t2;
      float dg0 = ng0 - g0, dg1 = ng1 - g1, dg2 = ng2 - g2;
      float Jd0 = Ji[0][0] * dg0 + Ji[0][1] * dg1 + Ji[0][2] * dg2;
      float Jd1 = Ji[1][0] * dg0 + Ji[1][1] * dg1 + Ji[1][2] * dg2;
      float Jd2 = Ji[2][0] * dg0 + Ji[2][1] * dg1 + Ji[2][2] * dg2;
      float denom = u0 * Jd0 + u1 * Jd1 + u2 * Jd2 + 1e-6f;
      float v0 = u0 * Ji[0][0] + u1 * Ji[1][0] + u2 * Ji[2][0];
      float v1 = u0 * Ji[0][1] + u1 * Ji[1][1] + u2 * Ji[2][1];
      float v2 = u0 * Ji[0][2] + u1 * Ji[1][2] + u2 * Ji[2][2];
      float idn = 1.0f / denom;
      float w0 = (u0 - Jd0) * idn, w1 = (u1 - Jd1) * idn, w2 = (u2 - Jd2) * idn;
      Ji[0][0] += w0 * v0; Ji[0][1] += w0 * v1; Ji[0][2] += w0 * v2;
      Ji[1][0] += w1 * v0; Ji[1][1] += w1 * v1; Ji[1][2] += w1 * v2;
      Ji[2][0] += w2 * v0; Ji[2][1] += w2 * v1; Ji[2][2] += w2 * v2;
      g0 = ng0; g1 = ng1; g2 = ng2;
      float err = sqrtf(g0 * g0 + g1 * g1 + g2 * g2);
      if (err < berr) { berr = err; bx0 = x0; bx1 = x1; bx2 = x2; }
    }
    __syncthreads();
  }

  if (half == 0) {
    out[(size_t)m * 3 + 0] = bx0;
    out[(size_t)m * 3 + 1] = bx1;
    out[(size_t)m * 3 + 2] = bx2;
    out[(size_t)M * 3 + m] = (berr < CVG_THRESHF) ? 1.0f : 0.0f;
  }
}

extern "C" void kernel_launch(void* const* d_in, const int* in_sizes, int n_in,
                              void* d_out, int out_size, void* d_ws, size_t ws_size,
                              hipStream_t stream) {
  const float* xd  = (const float*)d_in[0];
  const float* tfs = (const float*)d_in[1];
  const float* W1  = (const float*)d_in[2];
  const float* b1  = (const float*)d_in[3];
  const float* W2  = (const float*)d_in[4];
  const float* b2  = (const float*)d_in[5];
  const float* W3  = (const float*)d_in[6];
  const float* b3  = (const float*)d_in[7];
  int N = in_sizes[0] / 3;
  int M = N * NIB;
  int blocks = (M + BLK_PTS - 1) / BLK_PTS;
  deformer_broyden_kernel<<<blocks, NWAVE * 32, 0, stream>>>(
      xd, tfs, W1, b1, W2, b2, W3, b3, (float*)d_out, N);
}